// TestBackbone_87385404605171
// MI455X (gfx1250) — compile-verified
//
#include <hip/hip_runtime.h>

// ---------------------------------------------------------------------------
// MI455X / gfx1250 implementation of the dense voxel backbone.
// All 3D convolutions run as implicit GEMM on V_WMMA_F32_16X16X4_F32
// (full fp32 path through the matrix unit, matching reference numerics).
//
// v2: - weights repacked per-conv into [tap][ci/4][khalf][co*2] so the
//       A-fragment is one coalesced global_load_b64 with immediate offset
//     - kernel templated on <CIN,COUT>: ci-chunk loop fully unrolled,
//       loads clause/pipeline instead of s_wait_loadcnt 0 per WMMA
//     - halo handled by clamped address + 0/1 multiply (no EXEC juggling)
// ---------------------------------------------------------------------------

typedef __attribute__((ext_vector_type(2))) float v2f;
typedef __attribute__((ext_vector_type(8))) float v8f;

#define BN_EPS 1e-3f

// ---------------------------------------------------------------- fill -----
__global__ void bb_fill_kernel(float* __restrict__ p, size_t n, float v) {
  size_t i = (size_t)blockIdx.x * blockDim.x + threadIdx.x;
  if (i < n) p[i] = v;
}

// ------------------------------------------------------------- scatter -----
__global__ void bb_scatter_kernel(const float* __restrict__ vf,
                                  const int* __restrict__ coors,
                                  float* __restrict__ grid,
                                  float* __restrict__ mask,
                                  int M, int C, int D, int H, int W) {
  int i = blockIdx.x * blockDim.x + threadIdx.x;
  if (i >= M * C) return;
  int m = i / C;
  int c = i - m * C;
  int z = coors[m * 4 + 1];
  int y = coors[m * 4 + 2];
  int x = coors[m * 4 + 3];
  size_t sp = ((size_t)z * H + y) * W + x;
  grid[(size_t)c * D * H * W + sp] = vf[i];
  if (c == 0) mask[sp] = 1.0f;
}

// ------------------------------------------- fold bias+BN into scale/shift --
__global__ void bb_affine_kernel(const float* __restrict__ g,
                                 const float* __restrict__ b,
                                 const float* __restrict__ m,
                                 const float* __restrict__ v,
                                 const float* __restrict__ bias,
                                 float* __restrict__ scale,
                                 float* __restrict__ shift, int C) {
  int c = blockIdx.x * blockDim.x + threadIdx.x;
  if (c >= C) return;
  float sc = g[c] * rsqrtf(v[c] + BN_EPS);
  float bi = bias ? bias[c] : 0.0f;
  scale[c] = sc;
  shift[c] = (bi - m[c]) * sc + b[c];
}

// ------------------------------------------------- mask pooling ( >0 ) -----
__global__ void bb_maskpool_kernel(const float* __restrict__ mi,
                                   float* __restrict__ mo,
                                   int ID, int IH, int IW,
                                   int OD, int OH, int OW,
                                   int KD, int KH, int KW,
                                   int sd, int sh, int sw,
                                   int pd, int ph, int pw) {
  int i = blockIdx.x * blockDim.x + threadIdx.x;
  int total = OD * OH * OW;
  if (i >= total) return;
  int ow = i % OW;
  int t = i / OW;
  int oh = t % OH;
  int od = t / OH;
  float s = 0.0f;
  for (int kd = 0; kd < KD; ++kd) {
    int id = od * sd - pd + kd;
    if ((unsigned)id >= (unsigned)ID) continue;
    for (int kh = 0; kh < KH; ++kh) {
      int ih = oh * sh - ph + kh;
      if ((unsigned)ih >= (unsigned)IH) continue;
      for (int kw = 0; kw < KW; ++kw) {
        int iw = ow * sw - pw + kw;
        if ((unsigned)iw >= (unsigned)IW) continue;
        s += mi[((size_t)id * IH + ih) * IW + iw];
      }
    }
  }
  mo[i] = (s > 0.0f) ? 1.0f : 0.0f;
}

// ----------------------------------------------------- weight repack -------
// w:  [co][ci][tap]                       (OIDHW flattened)
// wr: ((tap*(Cin/4) + c4)*2 + h)*(2*Cout) + co*2 + j,  ci = 4*c4 + 2*h + j
// -> per WMMA chunk the A fragment for lane (n,khalf) is ONE b64 load,
//    coalesced across lanes, with a compile-time immediate offset.
__global__ void bb_wrepack_kernel(const float* __restrict__ w,
                                  float* __restrict__ wr,
                                  int Cout, int Cin, int tapsz) {
  int i = blockIdx.x * blockDim.x + threadIdx.x;
  int total = Cout * Cin * tapsz;
  if (i >= total) return;
  int tap = i % tapsz;
  int t = i / tapsz;
  int ci = t % Cin;
  int co = t / Cin;
  int c4 = ci >> 2;
  int r = ci & 3;
  int h = r >> 1;
  int j = r & 1;
  size_t o = ((size_t)(tap * (Cin >> 2) + c4) * 2 + h) * ((size_t)Cout * 2) +
             (size_t)co * 2 + j;
  wr[o] = w[i];
}

// ---------------------------------------------------------------------------
// Fused conv3d + (bias+BN folded) + mask-mul + residual + ReLU.
// One wave computes a 16 out-channel x 16 spatial (along W) tile.
//   A (16x4): lanes 0-15 M=lane, K=0,1; lanes 16-31 K=2,3   (one b64 load)
//   B (4x16): lanes 0-15 N=lane, K=0,1; lanes 16-31 K=2,3   (two b32 loads)
//   C/D:      lanes 0-15 -> M=v, lanes 16-31 -> M=v+8
// ---------------------------------------------------------------------------
template <int CIN, int COUT>
__global__ __launch_bounds__(32) void bb_conv3d_wmma_kernel(
    const float* __restrict__ x, const float* __restrict__ wr,
    const float* __restrict__ scale, const float* __restrict__ shift,
    const float* __restrict__ mask, const float* __restrict__ residual,
    float* __restrict__ y,
    int ID, int IH, int IW,
    int OD, int OH, int OW,
    int KD, int KH, int KW,
    int sd, int sh, int sw,
    int pd, int ph, int pw) {
  const int lane  = threadIdx.x;
  const int n     = lane & 15;   // spatial column / out-channel row in tile
  const int khalf = lane >> 4;   // K=0,1 (0) vs K=2,3 (1)

  const int tiles_w = OW >> 4;
  int t = blockIdx.x;
  const int tw = t % tiles_w; t /= tiles_w;
  const int oh = t % OH;      t /= OH;
  const int od = t % OD;
  const int co_base = (t / OD) << 4;

  const int ow  = (tw << 4) + n;
  const int id0 = od * sd - pd;
  const int ih0 = oh * sh - ph;
  const int iw0 = ow * sw - pw;

  const size_t IHW  = (size_t)IH * IW;
  const size_t IDHW = (size_t)ID * IHW;
  const size_t CHD4 = 4 * IDHW;            // B stride per ci-chunk

  // per-lane repacked-weight base (see bb_wrepack_kernel)
  const float* __restrict__ wlane =
      wr + (size_t)khalf * (2 * COUT) + (size_t)(co_base + n) * 2;
  constexpr int WTAP = CIN * COUT;         // floats per tap in wr

  v8f acc = {0.f, 0.f, 0.f, 0.f, 0.f, 0.f, 0.f, 0.f};

  int tap = 0;
  for (int kd = 0; kd < KD; ++kd) {
    const int id = id0 + kd;
    if ((unsigned)id >= (unsigned)ID) { tap += KH * KW; continue; }  // uniform
    for (int kh = 0; kh < KH; ++kh) {
      const int ih = ih0 + kh;
      if ((unsigned)ih >= (unsigned)IH) { tap += KW; continue; }     // uniform
      const float* __restrict__ row =
          x + (size_t)id * IHW + (size_t)ih * IW + (size_t)(2 * khalf) * IDHW;
      __builtin_prefetch(row + IW, 0, 1);  // global_prefetch_b8 (next row)
      for (int kw_ = 0; kw_ < KW; ++kw_, ++tap) {
        const int  iw  = iw0 + kw_;
        const bool ok  = (unsigned)iw < (unsigned)IW;  // per-lane halo
        const int  iwc = ok ? iw : 0;                  // clamped: load in-range
        const float okf = ok ? 1.0f : 0.0f;            // zero contribution OOB
        const float* __restrict__ xp = row + iwc;
        const float* __restrict__ wp = wlane + (size_t)tap * WTAP;
#pragma unroll
        for (int c4 = 0; c4 < CIN / 4; ++c4) {
          v2f a = *(const v2f*)(wp + c4 * (4 * COUT));  // immediate offset b64
          v2f b;
          b.x = xp[0]    * okf;
          b.y = xp[IDHW] * okf;
          xp += CHD4;
          acc = __builtin_amdgcn_wmma_f32_16x16x4_f32(
              false, a, false, b, (short)0, acc, false, false);
        }
      }
    }
  }

  // ------------------------------- epilogue -------------------------------
  const float  mv  = mask[((size_t)od * OH + oh) * OW + ow];
  const size_t OHW = (size_t)OH * OW;
  const int    cb  = co_base + khalf * 8;
  float scv[8], shv[8];
  *(float4*)(scv)     = *(const float4*)(scale + cb);
  *(float4*)(scv + 4) = *(const float4*)(scale + cb + 4);
  *(float4*)(shv)     = *(const float4*)(shift + cb);
  *(float4*)(shv + 4) = *(const float4*)(shift + cb + 4);
  size_t idx = ((size_t)cb * OD + od) * OHW + (size_t)oh * OW + ow;
  const size_t cstride = (size_t)OD * OHW;
#pragma unroll
  for (int v = 0; v < 8; ++v) {
    float r = acc[v] * scv[v] + shv[v];
    r *= mv;
    if (residual) r += residual[idx];
    y[idx] = fmaxf(r, 0.0f);
    idx += cstride;
  }
}

// ------------------------------------------------------------ host side ----
struct BNP { const float *g, *b, *m, *v; };
struct BLK { const float *w1, *b1; BNP bn1; const float *w2, *b2; BNP bn2; };
struct STG { const float *w; BNP bn; BLK b1, b2; };

static inline BNP getBN(void* const* d, int& i) {
  BNP r;
  r.g = (const float*)d[i++]; r.b = (const float*)d[i++];
  r.m = (const float*)d[i++]; r.v = (const float*)d[i++];
  return r;
}
static inline BLK getBLK(void* const* d, int& i) {
  BLK r;
  r.w1 = (const float*)d[i++]; r.b1 = (const float*)d[i++]; r.bn1 = getBN(d, i);
  r.w2 = (const float*)d[i++]; r.b2 = (const float*)d[i++]; r.bn2 = getBN(d, i);
  return r;
}
static inline STG getSTG(void* const* d, int& i) {
  STG r;
  r.w = (const float*)d[i++]; r.bn = getBN(d, i);
  r.b1 = getBLK(d, i); r.b2 = getBLK(d, i);
  return r;
}

// repack weights then launch the templated implicit-GEMM conv
template <int CIN, int COUT>
static inline void run_conv(const float* x, const float* w, float* WR,
                            const float* SC, const float* SH,
                            const float* mask, const float* res, float* y,
                            int ID, int IH, int IW, int OD, int OH, int OW,
                            int KD, int KH, int KW, int sd, int sh, int sw,
                            int pd, int ph, int pw, hipStream_t stream) {
  const int tapsz = KD * KH * KW;
  const int total = COUT * CIN * tapsz;
  bb_wrepack_kernel<<<(total + 255) / 256, 256, 0, stream>>>(w, WR, COUT, CIN,
                                                             tapsz);
  const int tiles = (COUT / 16) * OD * OH * (OW / 16);
  bb_conv3d_wmma_kernel<CIN, COUT><<<tiles, 32, 0, stream>>>(
      x, WR, SC, SH, mask, res, y, ID, IH, IW, OD, OH, OW, KD, KH, KW,
      sd, sh, sw, pd, ph, pw);
}

extern "C" void kernel_launch(void* const* d_in, const int* in_sizes, int n_in,
                              void* d_out, int out_size, void* d_ws, size_t ws_size,
                              hipStream_t stream) {
  int i = 0;
  const float* vf   = (const float*)d_in[i++];
  const float* w_in = (const float*)d_in[i++];
  BNP bn_in = getBN(d_in, i);
  BLK c1b1  = getBLK(d_in, i);
  BLK c1b2  = getBLK(d_in, i);
  STG s2    = getSTG(d_in, i);
  STG s3    = getSTG(d_in, i);
  STG s4    = getSTG(d_in, i);
  const float* w_ex = (const float*)d_in[i++];
  BNP bn_ex = getBN(d_in, i);
  const int* coors = (const int*)d_in[i++];
  // batch_size / input_shape inputs ignored (fixed problem size)

  const int M = in_sizes[0] / 16;       // 50000 voxels, 16 features each
  const int D = 25, H = 128, W = 128;

  // -------- workspace layout (floats) --------
  const size_t SZ1 = (size_t)16 * 25 * 128 * 128;  // 6,553,600 (largest tensor)
  float* G   = (float*)d_ws;            // grid, later ping-pong buffer
  float* T1  = G + SZ1;
  float* T2  = T1 + SZ1;
  float* MK1 = T2 + SZ1;                // 25*128*128 = 409600
  float* MK2 = MK1 + 409600;            // 13*64*64   = 53248
  float* MK3 = MK2 + 53248;             // 7*32*32    = 7168
  float* MK4 = MK3 + 7168;              // 3*16*16    = 768
  float* M5  = MK4 + 768;               // 1*16*16    = 256
  float* SC  = M5 + 256;                // per-conv scale (<=128)
  float* SH  = SC + 128;                // per-conv shift (<=128)
  float* WR  = SH + 128;                // repacked weights (<=128*128*27)

  // -------- output slices: (ret, x_conv1, x_conv2, x_conv3, x_conv4) --------
  float* OUT = (float*)d_out;
  float* RET = OUT;                     // 128*16*16       = 32768
  float* X1  = RET + 32768;             // 16*25*128*128   = 6553600
  float* X2  = X1 + 6553600;            // 32*13*64*64     = 1703936
  float* X3  = X2 + 1703936;            // 64*7*32*32      = 458752
  float* X4  = X3 + 458752;             // 128*3*16*16     = 98304

  auto fill = [&](float* p, size_t n) {
    bb_fill_kernel<<<(unsigned)((n + 255) / 256), 256, 0, stream>>>(p, n, 0.0f);
  };
  auto affine = [&](BNP bn, const float* bias, int C) {
    bb_affine_kernel<<<1, 128, 0, stream>>>(bn.g, bn.b, bn.m, bn.v, bias, SC, SH, C);
  };
  auto pool = [&](const float* mi, float* mo, int ID, int IH, int IW,
                  int OD, int OH, int OW, int KD, int KH, int KW,
                  int sd, int sh, int sw, int pd, int ph, int pw) {
    int total = OD * OH * OW;
    bb_maskpool_kernel<<<(total + 255) / 256, 256, 0, stream>>>(
        mi, mo, ID, IH, IW, OD, OH, OW, KD, KH, KW, sd, sh, sw, pd, ph, pw);
  };

  // -------- scatter voxel features into dense grid + mask --------
  fill(G, SZ1);
  fill(MK1, 409600);
  bb_scatter_kernel<<<(M * 16 + 255) / 256, 256, 0, stream>>>(
      vf, coors, G, MK1, M, 16, D, H, W);

  // -------- stage 1 @ 16ch, dims (25,128,128) --------
  affine(bn_in, nullptr, 16);
  run_conv<16,16>(G,  w_in,     WR, SC, SH, MK1, nullptr, T1, 25,128,128, 25,128,128, 3,3,3, 1,1,1, 1,1,1, stream);
  affine(c1b1.bn1, c1b1.b1, 16);
  run_conv<16,16>(T1, c1b1.w1,  WR, SC, SH, MK1, nullptr, T2, 25,128,128, 25,128,128, 3,3,3, 1,1,1, 1,1,1, stream);
  affine(c1b1.bn2, c1b1.b2, 16);
  run_conv<16,16>(T2, c1b1.w2,  WR, SC, SH, MK1, T1,      G,  25,128,128, 25,128,128, 3,3,3, 1,1,1, 1,1,1, stream);
  affine(c1b2.bn1, c1b2.b1, 16);
  run_conv<16,16>(G,  c1b2.w1,  WR, SC, SH, MK1, nullptr, T2, 25,128,128, 25,128,128, 3,3,3, 1,1,1, 1,1,1, stream);
  affine(c1b2.bn2, c1b2.b2, 16);
  run_conv<16,16>(T2, c1b2.w2,  WR, SC, SH, MK1, G,       X1, 25,128,128, 25,128,128, 3,3,3, 1,1,1, 1,1,1, stream);

  // -------- stage 2: 16 -> 32ch, (25,128,128) -> (13,64,64) --------
  pool(MK1, MK2, 25,128,128, 13,64,64, 3,3,3, 2,2,2, 1,1,1);
  affine(s2.bn, nullptr, 32);
  run_conv<16,32>(X1, s2.w,     WR, SC, SH, MK2, nullptr, T1, 25,128,128, 13,64,64, 3,3,3, 2,2,2, 1,1,1, stream);
  affine(s2.b1.bn1, s2.b1.b1, 32);
  run_conv<32,32>(T1, s2.b1.w1, WR, SC, SH, MK2, nullptr, T2, 13,64,64, 13,64,64, 3,3,3, 1,1,1, 1,1,1, stream);
  affine(s2.b1.bn2, s2.b1.b2, 32);
  run_conv<32,32>(T2, s2.b1.w2, WR, SC, SH, MK2, T1,      G,  13,64,64, 13,64,64, 3,3,3, 1,1,1, 1,1,1, stream);
  affine(s2.b2.bn1, s2.b2.b1, 32);
  run_conv<32,32>(G,  s2.b2.w1, WR, SC, SH, MK2, nullptr, T2, 13,64,64, 13,64,64, 3,3,3, 1,1,1, 1,1,1, stream);
  affine(s2.b2.bn2, s2.b2.b2, 32);
  run_conv<32,32>(T2, s2.b2.w2, WR, SC, SH, MK2, G,       X2, 13,64,64, 13,64,64, 3,3,3, 1,1,1, 1,1,1, stream);

  // -------- stage 3: 32 -> 64ch, (13,64,64) -> (7,32,32) --------
  pool(MK2, MK3, 13,64,64, 7,32,32, 3,3,3, 2,2,2, 1,1,1);
  affine(s3.bn, nullptr, 64);
  run_conv<32,64>(X2, s3.w,     WR, SC, SH, MK3, nullptr, T1, 13,64,64, 7,32,32, 3,3,3, 2,2,2, 1,1,1, stream);
  affine(s3.b1.bn1, s3.b1.b1, 64);
  run_conv<64,64>(T1, s3.b1.w1, WR, SC, SH, MK3, nullptr, T2, 7,32,32, 7,32,32, 3,3,3, 1,1,1, 1,1,1, stream);
  affine(s3.b1.bn2, s3.b1.b2, 64);
  run_conv<64,64>(T2, s3.b1.w2, WR, SC, SH, MK3, T1,      G,  7,32,32, 7,32,32, 3,3,3, 1,1,1, 1,1,1, stream);
  affine(s3.b2.bn1, s3.b2.b1, 64);
  run_conv<64,64>(G,  s3.b2.w1, WR, SC, SH, MK3, nullptr, T2, 7,32,32, 7,32,32, 3,3,3, 1,1,1, 1,1,1, stream);
  affine(s3.b2.bn2, s3.b2.b2, 64);
  run_conv<64,64>(T2, s3.b2.w2, WR, SC, SH, MK3, G,       X3, 7,32,32, 7,32,32, 3,3,3, 1,1,1, 1,1,1, stream);

  // -------- stage 4: 64 -> 128ch, (7,32,32) -> (3,16,16), pad (0,1,1) --------
  pool(MK3, MK4, 7,32,32, 3,16,16, 3,3,3, 2,2,2, 0,1,1);
  affine(s4.bn, nullptr, 128);
  run_conv<64,128>(X3, s4.w,      WR, SC, SH, MK4, nullptr, T1, 7,32,32, 3,16,16, 3,3,3, 2,2,2, 0,1,1, stream);
  affine(s4.b1.bn1, s4.b1.b1, 128);
  run_conv<128,128>(T1, s4.b1.w1, WR, SC, SH, MK4, nullptr, T2, 3,16,16, 3,16,16, 3,3,3, 1,1,1, 1,1,1, stream);
  affine(s4.b1.bn2, s4.b1.b2, 128);
  run_conv<128,128>(T2, s4.b1.w2, WR, SC, SH, MK4, T1,      G,  3,16,16, 3,16,16, 3,3,3, 1,1,1, 1,1,1, stream);
  affine(s4.b2.bn1, s4.b2.b1, 128);
  run_conv<128,128>(G,  s4.b2.w1, WR, SC, SH, MK4, nullptr, T2, 3,16,16, 3,16,16, 3,3,3, 1,1,1, 1,1,1, stream);
  affine(s4.b2.bn2, s4.b2.b2, 128);
  run_conv<128,128>(T2, s4.b2.w2, WR, SC, SH, MK4, G,       X4, 3,16,16, 3,16,16, 3,3,3, 1,1,1, 1,1,1, stream);

  // -------- final extra conv: (3,1,1) stride (2,1,1) pad 0 -> ret --------
  pool(MK4, M5, 3,16,16, 1,16,16, 3,1,1, 2,1,1, 0,0,0);
  affine(bn_ex, nullptr, 128);
  run_conv<128,128>(X4, w_ex, WR, SC, SH, M5, nullptr, RET, 3,16,16, 1,16,16, 3,1,1, 2,1,1, 0,0,0, stream);
}